// Loss_31267361915290
// MI455X (gfx1250) — compile-verified
//
#include <hip/hip_runtime.h>
#include <stdint.h>

#define EPS_F 0.1f
#define BB 4
#define CC 2048
#define VV 32000
#define ROWS (BB * (CC - 1))        // 8188
#define TPB 320                     // 10 waves of 32
#define F4_PER_ROW (VV / 4)         // 8000
#define ITERS (F4_PER_ROW / TPB)    // 25 exactly
#define NSTAGE 4                    // LDS ring depth (power of 2)

typedef __attribute__((ext_vector_type(2))) float v2f;
typedef __attribute__((ext_vector_type(8))) float v8f;

// ---- CDNA5 async global->LDS copy (per-lane, 16B each), non-temporal stream ----
__device__ __forceinline__ void async_cp_f4(uint32_t lds_addr, const float4* gptr) {
  asm volatile("global_load_async_to_lds_b128 %0, %1, off th:TH_LOAD_NT"
               :: "v"(lds_addr), "v"((uint64_t)(uintptr_t)gptr)
               : "memory");
}

template <int N>
__device__ __forceinline__ void wait_async() {
  asm volatile("s_wait_asynccnt %0" :: "n"(N) : "memory");
}

// ---- online-softmax combine: (m1,s1) (+) (m2,s2) ----
__device__ __forceinline__ void combine_ms(float& m, float& s, float om, float os) {
  float nm = fmaxf(m, om);
  s = s * __expf(m - nm) + os * __expf(om - nm);
  m = nm;
}

// ================= kernel 0: S = sum(token_histo) =================
__global__ void histo_sum_kernel(const float* __restrict__ histo, float* __restrict__ wsS) {
  __shared__ float red[8];
  float s = 0.f;
  for (int i = threadIdx.x; i < VV; i += 256) s += histo[i];
#pragma unroll
  for (int o = 16; o; o >>= 1) s += __shfl_xor(s, o, 32);
  if ((threadIdx.x & 31) == 0) red[threadIdx.x >> 5] = s;
  __syncthreads();
  if (threadIdx.x == 0) {
    float t = 0.f;
#pragma unroll
    for (int w = 0; w < 8; ++w) t += red[w];
    wsS[0] = t;
  }
}

// ================= main: one block per logits row =================
__global__ __launch_bounds__(TPB) void xent_row_kernel(
    const int* __restrict__ dec_input, const float* __restrict__ dec_output,
    const float* __restrict__ histo, const float* __restrict__ wsS,
    float* __restrict__ rowOut) {
  const int row = blockIdx.x;
  const int b = row / (CC - 1);
  const int c = row % (CC - 1);
  const float* rowPtr = dec_output + (size_t)(b * CC + c) * VV;
  const float4* rowF4 = (const float4*)rowPtr;
  const float4* histF4 = (const float4*)histo;
  const int tid = threadIdx.x;

  __shared__ float4 stage[NSTAGE][TPB];   // 20 KB ring
  __shared__ float rm[TPB / 32], rs[TPB / 32], rd[TPB / 32];

  // label logit (thread 0 only; loads issued early, consumed after the loop)
  float xl = 0.f;
  if (tid == 0) {
    int lbl = dec_input[b * CC + c + 1];
    xl = rowPtr[lbl];
  }

  // prologue: stages 0..2 in flight
#pragma unroll
  for (int p = 0; p < NSTAGE - 1; ++p) {
    async_cp_f4((uint32_t)(uintptr_t)&stage[p][tid], rowF4 + (p * TPB + tid));
  }

  float m = -__builtin_inff(), s = 0.f, d = 0.f;

  auto consume = [&](int i) {
    float4 x = stage[i & (NSTAGE - 1)][tid];
    float4 h = histF4[i * TPB + tid];
    float xs[4] = {x.x, x.y, x.z, x.w};
    float hs[4] = {h.x, h.y, h.z, h.w};
#pragma unroll
    for (int e = 0; e < 4; ++e) {
      float v = xs[e];
      d = fmaf(v, hs[e], d);
      if (v > m) { s = s * __expf(m - v) + 1.0f; m = v; }   // rare rescale path
      else       { s += __expf(v - m); }
    }
  };

  // steady state: keep NSTAGE async copies in flight per wave
  for (int i = 0; i + (NSTAGE - 1) < ITERS; ++i) {   // i = 0..21
    int j = i + (NSTAGE - 1);
    async_cp_f4((uint32_t)(uintptr_t)&stage[j & (NSTAGE - 1)][tid],
                rowF4 + (j * TPB + tid));
    wait_async<NSTAGE - 1>();   // oldest stage complete
    consume(i);
  }
  // drain
  wait_async<2>(); consume(ITERS - 3);
  wait_async<1>(); consume(ITERS - 2);
  wait_async<0>(); consume(ITERS - 1);

  // ---- wave32 reduction ----
#pragma unroll
  for (int o = 16; o; o >>= 1) {
    float om = __shfl_xor(m, o, 32);
    float os = __shfl_xor(s, o, 32);
    float od = __shfl_xor(d, o, 32);
    combine_ms(m, s, om, os);
    d += od;
  }
  const int wid = tid >> 5;
  if ((tid & 31) == 0) { rm[wid] = m; rs[wid] = s; rd[wid] = d; }
  __syncthreads();

  if (tid == 0) {
    float M = rm[0], S = rs[0], D = rd[0];
#pragma unroll
    for (int w = 1; w < TPB / 32; ++w) {
      combine_ms(M, S, rm[w], rs[w]);
      D += rd[w];
    }
    const float invH = 1.0f / wsS[0];            // 1 / sum(histo)
    const float logZ = M + __logf(S);
    rowOut[row] = logZ - (1.0f - EPS_F) * xl - EPS_F * (D * invH);
  }
}

// ================= finalize: deterministic mean over rows via WMMA =================
// Sum = A x ones(4x16) accumulated on the matrix pipe: D[m][n] = sum_k A[m][k] + C[m][n].
// Value->(M,K) placement is irrelevant for a pure sum; each V_WMMA_F32_16X16X4_F32
// folds 64 floats into the accumulator. One wave, EXEC all-ones in the WMMA section.
__global__ __launch_bounds__(32) void finalize_kernel(const float* __restrict__ rowOut,
                                                      float* __restrict__ out) {
  const int lane = threadIdx.x;

  v8f acc = {};
  v2f onesB;
  onesB.x = 1.0f; onesB.y = 1.0f;

  constexpr int NFULL = (ROWS / 64);        // 127 full 64-value chunks (8128 values)
  const float2* p2 = (const float2*)rowOut; // 2 values per lane per chunk

  for (int i = 0; i < NFULL; ++i) {
    float2 t = p2[i * 32 + lane];
    v2f a;
    a.x = t.x; a.y = t.y;
    acc = __builtin_amdgcn_wmma_f32_16x16x4_f32(
        /*neg_a=*/false, a, /*neg_b=*/false, onesB,
        /*c_mod=*/(short)0, acc, /*reuse_a=*/false, /*reuse_b=*/false);
  }

  // D column 0: rows 0..7 -> VGPRs 0..7 @ lane 0; rows 8..15 -> VGPRs 0..7 @ lane 16
  float t = acc[0] + acc[1] + acc[2] + acc[3] + acc[4] + acc[5] + acc[6] + acc[7];
  float wmmaSum = __shfl(t, 0, 32) + __shfl(t, 16, 32);

  // tail: remaining ROWS - 8128 = 60 values, masked lane-parallel adds
  float ex = 0.f;
  int i0 = NFULL * 64 + lane;
  if (i0 < ROWS) ex += rowOut[i0];
  int i1 = i0 + 32;
  if (i1 < ROWS) ex += rowOut[i1];
#pragma unroll
  for (int o = 16; o; o >>= 1) ex += __shfl_xor(ex, o, 32);

  if (lane == 0) out[0] = (wmmaSum + ex) / (float)ROWS;
}

extern "C" void kernel_launch(void* const* d_in, const int* in_sizes, int n_in,
                              void* d_out, int out_size, void* d_ws, size_t ws_size,
                              hipStream_t stream) {
  const int*   dec_input  = (const int*)d_in[0];    // (B, C) int32
  const float* dec_output = (const float*)d_in[1];  // (B, C, V) f32
  const float* histo      = (const float*)d_in[2];  // (V,) f32

  float* ws     = (float*)d_ws;
  float* wsS    = ws;        // ws[0]  = sum(histo)
  float* rowOut = ws + 16;   // 8188 per-row results (64B-aligned)

  histo_sum_kernel<<<1, 256, 0, stream>>>(histo, wsS);
  xent_row_kernel<<<ROWS, TPB, 0, stream>>>(dec_input, dec_output, histo, wsS, rowOut);
  finalize_kernel<<<1, 32, 0, stream>>>(rowOut, (float*)d_out);
}